// MemoryEfficientSplade_70763881168954
// MI455X (gfx1250) — compile-verified
//
#include <hip/hip_runtime.h>
#include <hip/hip_bf16.h>
#include <math.h>

#define BATCH   16
#define SEQ     256
#define VOCAB   30522
#define TILE_V  512
#define NWAVES  16          // 512 threads / wave32
#define DEPTH   8           // per-wave async pipeline depth (rows in flight)

// ---------------------------------------------------------------------------
// Kernel 1: values[b,v] = log1p(relu(max over unmasked s of logits[b,s,v]))
// Streaming max-reduction with a CDNA5 async global->LDS ring (depth 8) per
// wave. Each wave owns a 64-column stripe and only consumes LDS it loaded
// itself => no inter-wave barriers in the steady-state loop.
// ---------------------------------------------------------------------------
__global__ __launch_bounds__(512) void splade_max_kernel(
    const float* __restrict__ logits, const int* __restrict__ mask,
    float* __restrict__ out)
{
  __shared__ int      s_count;
  __shared__ unsigned s_rowoff[SEQ];             // byte offsets of unmasked rows
  __shared__ float    ring[NWAVES * DEPTH * 64]; // 32 KB: per-wave 8 slots x 256B

  const int b      = blockIdx.y;
  const int v_base = blockIdx.x * TILE_V;
  const int tid    = threadIdx.x;
  const int w      = tid >> 5;
  const int lane   = tid & 31;

  const unsigned ROWB = (unsigned)VOCAB * 4u;    // 122088 B per seq row

  // ---- Phase 0: compact byte-offsets of unmasked sequence positions ----
  if (tid == 0) s_count = 0;
  __syncthreads();
  if (tid < SEQ) {
    if (mask[b * SEQ + tid] != 0) {
      int p = atomicAdd(&s_count, 1);
      s_rowoff[p] = (unsigned)tid * ROWB;        // s * 122088 fits in 32 bits
    }
  }
  __syncthreads();
  const int N = s_count;

  // Lane owns column pair (v0, v0+1). Clamp load address for the vocab tail
  // (duplicate loads; results simply not stored).
  const int v0    = v_base + w * 64 + lane * 2;
  const int vload = (v0 <= VOCAB - 2) ? v0 : (VOCAB - 2);

  const unsigned long long g_lane =
      (unsigned long long)(const void*)logits +
      ((unsigned long long)b * SEQ * VOCAB + (unsigned long long)vload) * 4ull;

  // LDS byte address of this lane's 8B sub-slot in slot 0 of its wave's ring.
  // (low 32 bits of a generic pointer to LDS == ds-instruction byte address)
  const unsigned lds_lane =
      (unsigned)(unsigned long long)(const void*)&ring[(w * DEPTH) * 64] +
      (unsigned)(lane * 8);

  float acc0 = 0.0f, acc1 = 0.0f;

  if (N > 0) {
    // Prologue: fill the pipeline (row indices clamped to N-1; duplicate rows
    // are a no-op under max).
    #pragma unroll
    for (int j = 0; j < DEPTH; ++j) {
      const int ridx = (j < N) ? j : (N - 1);
      const unsigned long long ga = g_lane + (unsigned long long)s_rowoff[ridx];
      const unsigned la = lds_lane + (unsigned)(j * 256);
      asm volatile("global_load_async_to_lds_b64 %0, %1, off"
                   :: "v"(la), "v"(ga) : "memory");
    }
    // Steady state: async loads complete in order, so asynccnt <= DEPTH-1
    // guarantees the oldest slot (row i) has landed in LDS for this wave.
    for (int i = 0; i < N; ++i) {
      asm volatile("s_wait_asynccnt %0" :: "i"(DEPTH - 1) : "memory");
      const int slot = i & (DEPTH - 1);
      const float2 xv =
          *(const float2*)&ring[(w * DEPTH + slot) * 64 + lane * 2];
      acc0 = fmaxf(acc0, xv.x);
      acc1 = fmaxf(acc1, xv.y);
      // Pin ordering: force the DS read to drain into the accumulators before
      // the refill of the same slot can issue (volatile asm order preserved).
      asm volatile("" : "+v"(acc0), "+v"(acc1));
      // Refill this slot with row i+DEPTH (clamped).
      int nr = i + DEPTH; if (nr > N - 1) nr = N - 1;
      const unsigned long long ga = g_lane + (unsigned long long)s_rowoff[nr];
      const unsigned la = lds_lane + (unsigned)(slot * 256);
      asm volatile("global_load_async_to_lds_b64 %0, %1, off"
                   :: "v"(la), "v"(ga) : "memory");
    }
    asm volatile("s_wait_asynccnt 0" ::: "memory");
  }

  if (v0 <= VOCAB - 2) {
    float2 r; r.x = log1pf(acc0); r.y = log1pf(acc1);
    *(float2*)(out + (size_t)b * VOCAB + v0) = r;   // 8B aligned (v0 even)
  }
}

// ---------------------------------------------------------------------------
// Kernel 2: per batch row, exact radix-select (4 x 8-bit passes on float bits;
// all values >= 0 so IEEE order == unsigned order) -> threshold = k-th largest
// bit pattern, then zero out values < threshold (ties kept, matching ref).
// ---------------------------------------------------------------------------
__global__ __launch_bounds__(1024) void topk_threshold_kernel(
    float* __restrict__ vals, const int* __restrict__ topk_ptr)
{
  const int b = blockIdx.x;
  float* row = vals + (size_t)b * VOCAB;

  __shared__ unsigned hist[256];
  __shared__ unsigned sh_pref, sh_kth;

  unsigned k = (unsigned)topk_ptr[0];
  if (k < 1u) k = 1u;
  if (k > (unsigned)VOCAB) k = (unsigned)VOCAB;

  unsigned prefix = 0u, msk = 0u, kth = k;

  for (int shift = 24; shift >= 0; shift -= 8) {
    if (threadIdx.x < 256) hist[threadIdx.x] = 0u;
    __syncthreads();
    for (int i = threadIdx.x; i < VOCAB; i += blockDim.x) {
      const unsigned key = __float_as_uint(row[i]);
      if ((key & msk) == prefix)
        atomicAdd(&hist[(key >> shift) & 0xFF], 1u);
    }
    __syncthreads();
    if (threadIdx.x == 0) {
      unsigned cum = 0u;
      for (int bin = 255; bin >= 0; --bin) {
        const unsigned c = hist[bin];
        if (cum + c >= kth) {
          sh_pref = prefix | ((unsigned)bin << shift);
          sh_kth  = kth - cum;
          break;
        }
        cum += c;
      }
    }
    __syncthreads();
    prefix = sh_pref;
    kth    = sh_kth;
    msk   |= (0xFFu << shift);
  }

  const float thr = __uint_as_float(prefix);
  for (int i = threadIdx.x; i < VOCAB; i += blockDim.x) {
    const float v = row[i];
    row[i] = (v >= thr) ? v : 0.0f;
  }
}

// ---------------------------------------------------------------------------
extern "C" void kernel_launch(void* const* d_in, const int* in_sizes, int n_in,
                              void* d_out, int out_size, void* d_ws, size_t ws_size,
                              hipStream_t stream) {
  (void)in_sizes; (void)n_in; (void)out_size; (void)d_ws; (void)ws_size;
  const float* logits = (const float*)d_in[0];
  const int*   mask   = (const int*)d_in[1];
  const int*   topk   = (const int*)d_in[2];
  float*       out    = (float*)d_out;

  dim3 grid1((VOCAB + TILE_V - 1) / TILE_V, BATCH);   // 60 x 16 = 960 blocks
  splade_max_kernel<<<grid1, 512, 0, stream>>>(logits, mask, out);

  topk_threshold_kernel<<<BATCH, 1024, 0, stream>>>(out, topk);
}